// myLstm_24026047054718
// MI455X (gfx1250) — compile-verified
//
#include <hip/hip_runtime.h>
#include <math.h>

typedef __attribute__((ext_vector_type(2))) float v2f;
typedef __attribute__((ext_vector_type(8))) float v8f;

#define BS   64
#define SEQ  1024
#define IND  512
#define HS   512
#define ZD   2048  // 4*HS, gate order: i, f, c(g), o

// ---- workspace layout (float offsets) ----
#define XPROJ_OFF 0ull
#define XPROJ_SZ  ((size_t)SEQ * BS * ZD)          // 134217728 floats (512 MB)
#define WT_OFF    (XPROJ_OFF + XPROJ_SZ)
#define WT_SZ     ((size_t)ZD * IND)               // Wt[z][k] = W[k][z]
#define UT_OFF    (WT_OFF + WT_SZ)
#define UT_SZ     ((size_t)ZD * HS)                // Ut[z][k] = U[k][z]
#define BC_OFF    (UT_OFF + UT_SZ)
#define BC_SZ     ((size_t)ZD)
#define H_OFF     (BC_OFF + BC_SZ)
#define H_SZ      ((size_t)2 * BS * HS)            // double-buffered h
#define CNT_OFF   (H_OFF + H_SZ)                   // 8 floats: barrier cnt/gen

// ---------------------------------------------------------------------------
// Kernel 0: pack weights (transpose W,U; concat b; zero h + barrier counters)
// ---------------------------------------------------------------------------
__global__ void lstm_pack(const float* __restrict__ Wi, const float* __restrict__ Ui, const float* __restrict__ bi,
                          const float* __restrict__ Wf, const float* __restrict__ Uf, const float* __restrict__ bf,
                          const float* __restrict__ Wc, const float* __restrict__ Uc, const float* __restrict__ bc,
                          const float* __restrict__ Wo, const float* __restrict__ Uo, const float* __restrict__ bo,
                          float* __restrict__ ws) {
  size_t i = (size_t)blockIdx.x * blockDim.x + threadIdx.x;
  const float* Ws[4] = {Wi, Wf, Wc, Wo};
  const float* Us[4] = {Ui, Uf, Uc, Uo};
  const float* Bs[4] = {bi, bf, bc, bo};
  if (i < WT_SZ) {
    int z = (int)(i / IND), k = (int)(i % IND);
    int g = z >> 9, j = z & 511;
    ws[WT_OFF + i] = Ws[g][(size_t)k * HS + j];
  } else if (i < WT_SZ + UT_SZ) {
    size_t q = i - WT_SZ;
    int z = (int)(q / HS), k = (int)(q % HS);
    int g = z >> 9, j = z & 511;
    ws[UT_OFF + q] = Us[g][(size_t)k * HS + j];
  } else if (i < WT_SZ + UT_SZ + BC_SZ) {
    size_t q = i - WT_SZ - UT_SZ;
    int g = (int)(q >> 9), j = (int)(q & 511);
    ws[BC_OFF + q] = Bs[g][j];
  } else if (i < WT_SZ + UT_SZ + BC_SZ + H_SZ + 8) {
    ws[H_OFF + (i - (WT_SZ + UT_SZ + BC_SZ))] = 0.0f;  // h buffers + counters
  }
}

// ---------------------------------------------------------------------------
// Kernel 1: x_proj[s][b][z] = x[b][s][:] @ W[:, z] + bcat[z]
// One wave per 32(M) x 64(N) block: 2 A fragments x 4 B fragments = 8 accs.
// Halves L2 B-traffic vs 16xN blocking; 8 WMMAs per 6 8-byte loads.
// M = s*64 + b  (32-row blocks never cross the 64-row batch boundary).
// ---------------------------------------------------------------------------
__global__ __launch_bounds__(256) void lstm_xproj(const float* __restrict__ x,
                                                  float* __restrict__ ws) {
  const float* Wt   = ws + WT_OFF;
  const float* bcat = ws + BC_OFF;
  float*       xp   = ws + XPROJ_OFF;

  const int wid  = threadIdx.x >> 5;
  const int lane = threadIdx.x & 31;
  const int hi   = lane >> 4;        // K half selector (ISA A/B layout)
  const int lr   = lane & 15;

  const int job = blockIdx.x * 8 + wid;   // 65536 wave-jobs
  const int mp  = job >> 5;               // 0..2047 (32-row M blocks)
  const int n0  = (job & 31) * 64;        // N strip of 4 tiles
  const int M0  = mp * 32;
  const int s   = M0 >> 6;
  const int b0  = M0 & 63;                // 0 or 32

  v8f acc[2][4];
#pragma unroll
  for (int t = 0; t < 4; ++t) {
    const float bv = bcat[n0 + t * 16 + lr];
#pragma unroll
    for (int v = 0; v < 8; ++v) { acc[0][t][v] = bv; acc[1][t][v] = bv; }
  }

  const float* ap0 = x  + ((size_t)(b0 + lr) * SEQ + s) * IND + 2 * hi;
  const float* ap1 = ap0 + (size_t)16 * SEQ * IND;
  const float* bp  = Wt + (size_t)(n0 + lr) * IND + 2 * hi;

  for (int k = 0; k < IND; k += 4) {
    const v2f a0 = *(const v2f*)(ap0 + k);
    const v2f a1 = *(const v2f*)(ap1 + k);
#pragma unroll
    for (int t = 0; t < 4; ++t) {
      const v2f bb = *(const v2f*)(bp + (size_t)t * 16 * IND + k);
      acc[0][t] = __builtin_amdgcn_wmma_f32_16x16x4_f32(false, a0, false, bb,
                                                        (short)0, acc[0][t], false, false);
      acc[1][t] = __builtin_amdgcn_wmma_f32_16x16x4_f32(false, a1, false, bb,
                                                        (short)0, acc[1][t], false, false);
    }
  }

#pragma unroll
  for (int m = 0; m < 2; ++m) {
    const size_t rowbase = ((size_t)s * BS + b0 + 16 * m + 8 * hi) * ZD;
#pragma unroll
    for (int t = 0; t < 4; ++t)
#pragma unroll
      for (int v = 0; v < 8; ++v)
        xp[rowbase + (size_t)v * ZD + n0 + t * 16 + lr] = acc[m][t][v];
  }
}

// ---------------------------------------------------------------------------
// Kernel 2: persistent recurrence. 128 blocks x 128 threads (4 waves).
// Block = (batch tile mt, hidden tile nt); wave = gate. Grid barrier per step.
// ---------------------------------------------------------------------------
__global__ __launch_bounds__(128) void lstm_recur(float* __restrict__ ws,
                                                  float* __restrict__ out) {
  const float* Ut = ws + UT_OFF;
  const float* xp = ws + XPROJ_OFF;
  float*       hb = ws + H_OFF;                 // [2][64][512]
  unsigned*    cnt = (unsigned*)(ws + CNT_OFF);
  unsigned*    gen = cnt + 1;

  __shared__ float ldsZ[4][16][16];
  __shared__ float ldsC[16][16];

  const int tid  = threadIdx.x;
  const int g    = tid >> 5;        // gate = wave id
  const int lane = tid & 31;
  const int hi   = lane >> 4;
  const int lr   = lane & 15;
  const int wg   = blockIdx.x;
  const int mt   = wg >> 5, nt = wg & 31;
  const int m0   = mt * 16, n0 = nt * 16;
  const int zb   = g * HS + n0;

  ldsC[tid >> 4][tid & 15] = 0.0f;
  ldsC[(tid + 128) >> 4][tid & 15] = 0.0f;
  __syncthreads();

  const float* bp = Ut + (size_t)(zb + lr) * HS + 2 * hi;

  for (int s = 0; s < SEQ; ++s) {
    const float* hread  = hb + (size_t)(s & 1) * BS * HS;
    float*       hwrite = hb + (size_t)((s + 1) & 1) * BS * HS;

    // C init from precomputed x-projection; second chain starts at zero.
    v8f acc0, acc1;
    const size_t xb = ((size_t)s * BS + m0 + 8 * hi) * ZD + zb + lr;
#pragma unroll
    for (int v = 0; v < 8; ++v) { acc0[v] = xp[xb + (size_t)v * ZD]; acc1[v] = 0.0f; }

    const float* ap = hread + (size_t)(m0 + lr) * HS + 2 * hi;
    for (int k = 0; k < HS; k += 8) {
      const v2f a0 = *(const v2f*)(ap + k);
      const v2f b0 = *(const v2f*)(bp + k);
      const v2f a1 = *(const v2f*)(ap + k + 4);
      const v2f b1 = *(const v2f*)(bp + k + 4);
      acc0 = __builtin_amdgcn_wmma_f32_16x16x4_f32(false, a0, false, b0,
                                                   (short)0, acc0, false, false);
      acc1 = __builtin_amdgcn_wmma_f32_16x16x4_f32(false, a1, false, b1,
                                                   (short)0, acc1, false, false);
    }
#pragma unroll
    for (int v = 0; v < 8; ++v) ldsZ[g][v + 8 * hi][lr] = acc0[v] + acc1[v];
    __syncthreads();

    // gate math: each thread owns 2 of the 256 (b, j) cells
#pragma unroll
    for (int e = tid; e < 256; e += 128) {
      const int r = e >> 4, cc = e & 15;
      const float zi = fmaxf(ldsZ[0][r][cc], 0.0f);
      const float zf = fmaxf(ldsZ[1][r][cc], 0.0f);
      const float zg = tanhf(ldsZ[2][r][cc]);
      const float zo = 1.0f / (1.0f + __expf(-ldsZ[3][r][cc]));
      const float cv = zf * ldsC[r][cc] + zi * zg;
      ldsC[r][cc] = cv;
      const float hv = zo * tanhf(cv);
      hwrite[(size_t)(m0 + r) * HS + n0 + cc] = hv;
      out[((size_t)(m0 + r) * SEQ + s) * HS + n0 + cc] = hv;
    }

    // device-wide barrier (sense via generation counter)
    __threadfence();
    __syncthreads();
    if (tid == 0) {
      const unsigned my = __hip_atomic_load(gen, __ATOMIC_RELAXED, __HIP_MEMORY_SCOPE_AGENT);
      const unsigned prev = __hip_atomic_fetch_add(cnt, 1u, __ATOMIC_ACQ_REL, __HIP_MEMORY_SCOPE_AGENT);
      if (prev == 127u) {
        __hip_atomic_store(cnt, 0u, __ATOMIC_RELAXED, __HIP_MEMORY_SCOPE_AGENT);
        __hip_atomic_fetch_add(gen, 1u, __ATOMIC_RELEASE, __HIP_MEMORY_SCOPE_AGENT);
      } else {
        while (__hip_atomic_load(gen, __ATOMIC_ACQUIRE, __HIP_MEMORY_SCOPE_AGENT) == my) {
          __builtin_amdgcn_s_sleep(1);
        }
      }
    }
    __syncthreads();
  }
}

// ---------------------------------------------------------------------------
extern "C" void kernel_launch(void* const* d_in, const int* in_sizes, int n_in,
                              void* d_out, int out_size, void* d_ws, size_t ws_size,
                              hipStream_t stream) {
  const float* x = (const float*)d_in[0];
  float* ws  = (float*)d_ws;
  float* out = (float*)d_out;

  // d_in order: x, W_i,U_i,b_i, W_f,U_f,b_f, W_c,U_c,b_c, W_o,U_o,b_o
  const size_t packTotal = WT_SZ + UT_SZ + BC_SZ + H_SZ + 8;
  const int packBlocks = (int)((packTotal + 255) / 256);
  lstm_pack<<<packBlocks, 256, 0, stream>>>(
      (const float*)d_in[1], (const float*)d_in[2], (const float*)d_in[3],
      (const float*)d_in[4], (const float*)d_in[5], (const float*)d_in[6],
      (const float*)d_in[7], (const float*)d_in[8], (const float*)d_in[9],
      (const float*)d_in[10], (const float*)d_in[11], (const float*)d_in[12],
      ws);

  // 65536 wave-jobs / 8 waves per block
  lstm_xproj<<<8192, 256, 0, stream>>>(x, ws);

  // persistent recurrence: exactly 128 resident workgroups
  lstm_recur<<<128, 128, 0, stream>>>(ws, out);
}